// QuadrupolePredictorEGNN_57707180589678
// MI455X (gfx1250) — compile-verified
//
#include <hip/hip_runtime.h>
#include <hip/hip_bf16.h>

// ---------------- constants (match reference) ----------------
#define BB    8
#define NN    128
#define DIM   128
#define DEPTH 3
#define MD    16
#define E2    514          // EDGE_IN*2
#define FSTR  516          // padded column stride for Fi/Fj in workspace
#define NROW  (BB*NN)      // 1024 node rows
#define JSPL  4            // j-range splits in edge pass (parallelism)

typedef float v2f __attribute__((ext_vector_type(2)));
typedef float v8f __attribute__((ext_vector_type(8)));

__device__ __forceinline__ v8f wmma4(v2f a, v2f b, v8f c) {
    // V_WMMA_F32_16X16X4_F32 : D = A(16x4) * B(4x16) + C(16x16), all fp32
    return __builtin_amdgcn_wmma_f32_16x16x4_f32(
        false, a, false, b, (short)0, c, false, false);
}
__device__ __forceinline__ float silu_f(float x) {
    return x / (1.0f + __expf(-x));
}

// ---------------- kernel 0: embedding + coords copy ----------------
__global__ void k_embed(const int* __restrict__ ids,
                        const float* __restrict__ coors_in,
                        const float* __restrict__ tok_emb,
                        float* __restrict__ fOut, float* __restrict__ cOut) {
    int row = blockIdx.x;          // 0..1023
    int d   = threadIdx.x;         // 0..127
    int id  = ids[row];
    fOut[row * DIM + d] = tok_emb[id * DIM + d];
    if (d < 3) cOut[row * 3 + d] = coors_in[row * 3 + d];
}

// ---------------- kernel 1: node pre-projection (Fi / Fj) ----------------
// Fi = feats @ ew1[0:128,:] + eb1 ; Fj = feats @ ew1[128:256,:]
// Pad columns 514..515 are written as exact zeros so downstream loops need no guards.
// grid (64 row-tiles, 33 col-tiles, 2 halves), block = 32 (one wave)
__global__ void k_node_pre(const float* __restrict__ feats,
                           const float* __restrict__ ew1l,   // 257x514 slab
                           const float* __restrict__ eb1l,   // 514
                           float* __restrict__ Fi, float* __restrict__ Fj) {
    const int lane = threadIdx.x;
    const int half = lane >> 4, rc = lane & 15;
    const int row0 = blockIdx.x * 16;
    const int col  = blockIdx.y * 16 + rc;
    const int hsel = blockIdx.z;
    const float* W = ew1l + hsel * 128 * E2;
    const float* arow = feats + (row0 + rc) * DIM;

    // clamp column for loads: out-of-range lanes fetch col 0; their D column
    // is discarded (we store 0 there), and WMMA columns are independent.
    const bool cok  = (col < E2);
    const int  colc = cok ? col : 0;

    v8f acc = {0.f,0.f,0.f,0.f,0.f,0.f,0.f,0.f};
    for (int k = 0; k < DIM; k += 4) {
        int d0 = k + 2 * half;
        v2f a; a.x = arow[d0];              a.y = arow[d0 + 1];
        v2f b; b.x = W[d0 * E2 + colc];     b.y = W[(d0 + 1) * E2 + colc];
        acc = wmma4(a, b, acc);
    }
    if (col < FSTR) {
        float bias = (hsel == 0 && cok) ? eb1l[col] : 0.f;
        float* out = hsel ? Fj : Fi;
        for (int v = 0; v < 8; ++v) {
            int row = row0 + 8 * half + v;
            out[row * FSTR + col] = cok ? (acc[v] + bias) : 0.f;
        }
    }
}

// ---------------- kernel 2: fused edge pass ----------------
// block = (b, i-tile of 16) x j-split; 8 waves, wave w handles 4 j's.
// Writes deterministic partial m_i / coord-delta slabs (combined later).
__global__ void __launch_bounds__(256)
k_edge(const float* __restrict__ Fi, const float* __restrict__ Fj,
       const float* __restrict__ cin,
       const float* __restrict__ wd,    // ew1 row 256 (514)
       const float* __restrict__ w2g,   // ew2 slab (514x16)
       const float* __restrict__ eb2l,  // 16
       const float* __restrict__ cw1l,  // 16x64
       const float* __restrict__ cb1l,  // 64
       const float* __restrict__ cw2l,  // 64
       const float* __restrict__ cb2l,  // 1
       float* __restrict__ mIp,         // out: JSPL x 1024x16 partials
       float* __restrict__ cPp) {       // out: JSPL x 1024x3  partial deltas
    __shared__ float sW2[FSTR * MD];    // ew2, zero-padded rows 514..515 (33KB)
    __shared__ float sCi[16][4];
    __shared__ float sWd[FSTR];
    __shared__ float sCw1[MD * 64];
    __shared__ float sCb1[64];
    __shared__ float sCw2[64];
    __shared__ float sEb2[MD];
    __shared__ float sCb2;
    __shared__ float sDist[8][16];
    __shared__ float sRel[8][16][4];
    __shared__ float sM[8][16][MD];
    __shared__ float sCwE[8][16];
    __shared__ float sRed[8][32][8];
    __shared__ float sCacc[8][16][4];

    const int tid  = threadIdx.x;
    const int wave = tid >> 5;
    const int lane = tid & 31;
    const int half = lane >> 4, rc = lane & 15;
    const int b     = blockIdx.x >> 3;
    const int itile = blockIdx.x & 7;
    const int js    = blockIdx.y;           // 0..JSPL-1
    const int node0 = b * NN + itile * 16;

    // ---- stage constants (pads -> exact zero) ----
    for (int i = tid; i < FSTR * MD; i += 256) sW2[i] = (i < E2 * MD) ? w2g[i] : 0.f;
    for (int d = tid; d < FSTR; d += 256)      sWd[d] = (d < E2) ? wd[d] : 0.f;
    for (int i = tid; i < MD * 64; i += 256)   sCw1[i] = cw1l[i];
    if (tid < 64) { sCb1[tid] = cb1l[tid]; sCw2[tid] = cw2l[tid]; }
    if (tid < MD) sEb2[tid] = eb2l[tid];
    if (tid < 48) sCi[tid / 3][tid % 3] = cin[(node0 + tid / 3) * 3 + tid % 3];
    if (tid == 0) sCb2 = cb2l[0];
    __syncthreads();

    const float* fiRow = Fi + (size_t)(node0 + rc) * FSTR;
    v8f macc = {0.f,0.f,0.f,0.f,0.f,0.f,0.f,0.f};
    float cax = 0.f, cay = 0.f, caz = 0.f;

    for (int jj = 0; jj < NN / (8 * JSPL); ++jj) {   // 4 iterations
        int j = js * (NN / JSPL) + wave * (NN / (8 * JSPL)) + jj;
        if (lane < 16) {   // rel & dist for the 16 i's of this tile
            float cjx = cin[(b * NN + j) * 3 + 0];
            float cjy = cin[(b * NN + j) * 3 + 1];
            float cjz = cin[(b * NN + j) * 3 + 2];
            float rx = sCi[lane][0] - cjx;
            float ry = sCi[lane][1] - cjy;
            float rz = sCi[lane][2] - cjz;
            sRel[wave][lane][0] = rx; sRel[wave][lane][1] = ry; sRel[wave][lane][2] = rz;
            sDist[wave][lane] = rx * rx + ry * ry + rz * rz;
        }
        __syncthreads();

        // ---- GEMM2: m = silu( h(514) @ ew2(514x16) + eb2 ), h built on the fly.
        // Branch-free: Fi/Fj/sWd/sW2 pads are exact zeros -> h(pad)=silu(0)=0.
        const float* fjRow = Fj + (size_t)(b * NN + j) * FSTR;
        float distr = sDist[wave][rc];
        v8f acc = {0.f,0.f,0.f,0.f,0.f,0.f,0.f,0.f};
        for (int k = 0; k < E2; k += 4) {
            int d0 = k + 2 * half;
            float h0 = silu_f(fiRow[d0]     + fjRow[d0]     + distr * sWd[d0]);
            float h1 = silu_f(fiRow[d0 + 1] + fjRow[d0 + 1] + distr * sWd[d0 + 1]);
            v2f a;  a.x = h0;                  a.y = h1;
            v2f bf; bf.x = sW2[d0 * MD + rc];  bf.y = sW2[(d0 + 1) * MD + rc];
            acc = wmma4(a, bf, acc);
        }
        for (int v = 0; v < 8; ++v) {
            float m = silu_f(acc[v] + sEb2[rc]);
            macc[v] += m;                       // m_i partial (fragment layout)
            sM[wave][8 * half + v][rc] = m;     // stage for coord MLP
        }
        __syncthreads();

        // ---- coord MLP: t = silu(m@cw1+cb1) ; cw = t@cw2 + cb2 ----
        float pv[8];
        for (int v = 0; v < 8; ++v) pv[v] = 0.f;
        for (int ct = 0; ct < 4; ++ct) {
            v8f a2c = {0.f,0.f,0.f,0.f,0.f,0.f,0.f,0.f};
            for (int k2 = 0; k2 < MD; k2 += 4) {
                int d0 = k2 + 2 * half;
                v2f a2; a2.x = sM[wave][rc][d0]; a2.y = sM[wave][rc][d0 + 1];
                v2f b2;
                b2.x = sCw1[d0 * 64 + ct * 16 + rc];
                b2.y = sCw1[(d0 + 1) * 64 + ct * 16 + rc];
                a2c = wmma4(a2, b2, a2c);
            }
            float w2c = sCw2[ct * 16 + rc];
            float bc  = sCb1[ct * 16 + rc];
            for (int v = 0; v < 8; ++v)
                pv[v] += silu_f(a2c[v] + bc) * w2c;
        }
        // reduce the 16-lane N dimension inside each half (lanes 0-15 / 16-31)
        for (int m = 1; m < 16; m <<= 1)
            for (int v = 0; v < 8; ++v)
                pv[v] += __shfl_xor(pv[v], m, 32);
        if ((lane & 15) == 0)
            for (int v = 0; v < 8; ++v)
                sCwE[wave][8 * half + v] = pv[v] + sCb2;
        __syncthreads();

        if (lane < 16) {   // accumulate cw * rel for this tile's 16 i's
            float cwv = sCwE[wave][lane];
            cax += cwv * sRel[wave][lane][0];
            cay += cwv * sRel[wave][lane][1];
            caz += cwv * sRel[wave][lane][2];
        }
        __syncthreads();   // protect sRel/sDist WAR for next j
    }

    // ---- cross-wave combine (deterministic, no atomics) ----
    for (int v = 0; v < 8; ++v) sRed[wave][lane][v] = macc[v];
    if (lane < 16) {
        sCacc[wave][lane][0] = cax; sCacc[wave][lane][1] = cay; sCacc[wave][lane][2] = caz;
    }
    __syncthreads();
    if (wave == 0) {
        float* mSlab = mIp + (size_t)js * NROW * MD;
        for (int v = 0; v < 8; ++v) {
            float tot = 0.f;
            for (int w = 0; w < 8; ++w) tot += sRed[w][lane][v];
            int row = 8 * half + v;
            mSlab[(node0 + row) * MD + rc] = tot;
        }
        if (lane < 16) {
            float* cSlab = cPp + (size_t)js * NROW * 3;
            for (int c = 0; c < 3; ++c) {
                float s = 0.f;
                for (int w = 0; w < 8; ++w) s += sCacc[w][lane][c];
                cSlab[(node0 + lane) * 3 + c] = s;
            }
        }
    }
}

// ---------------- kernel 2b: combine j-split partials ----------------
__global__ void k_combine(const float* __restrict__ mIp, const float* __restrict__ cPp,
                          const float* __restrict__ cin,
                          float* __restrict__ mI, float* __restrict__ cout) {
    int idx = blockIdx.x * 256 + threadIdx.x;     // grid 64 -> 16384 threads
    if (idx < NROW * MD) {
        float s = 0.f;
        for (int p = 0; p < JSPL; ++p) s += mIp[(size_t)p * NROW * MD + idx];
        mI[idx] = s;
    }
    if (idx < NROW * 3) {
        float s = cin[idx];
        for (int p = 0; p < JSPL; ++p) s += cPp[(size_t)p * NROW * 3 + idx];
        cout[idx] = s;
    }
}

// ---------------- kernel 3: node update ----------------
// feats' = silu([LN(f)|m_i] @ nw1 + nb1) @ nw2 + nb2 + f
__global__ void __launch_bounds__(256)
k_node_upd(const float* __restrict__ fin, const float* __restrict__ mI,
           const float* __restrict__ lng, const float* __restrict__ lnb,
           const float* __restrict__ nw1g, const float* __restrict__ nb1g,
           const float* __restrict__ nw2g, const float* __restrict__ nb2g,
           float* __restrict__ fout) {
    __shared__ float sF[16][DIM];
    __shared__ float sX[16][148];
    __shared__ float sY[16][256];

    const int tid  = threadIdx.x;
    const int wave = tid >> 5;
    const int lane = tid & 31;
    const int half = lane >> 4, rc = lane & 15;
    const int row0 = blockIdx.x * 16;

    for (int i = tid; i < 16 * DIM; i += 256)
        sF[i >> 7][i & 127] = fin[(row0 + (i >> 7)) * DIM + (i & 127)];
    __syncthreads();

    if (tid < 16) {        // layernorm, one row per thread (cheap vs GEMMs)
        int r = tid;
        float mu = 0.f;
        for (int d = 0; d < DIM; ++d) mu += sF[r][d];
        mu *= (1.f / DIM);
        float var = 0.f;
        for (int d = 0; d < DIM; ++d) { float dv = sF[r][d] - mu; var += dv * dv; }
        var *= (1.f / DIM);
        float rs = rsqrtf(var + 1e-5f);
        for (int d = 0; d < DIM; ++d)
            sX[r][d] = (sF[r][d] - mu) * rs * lng[d] + lnb[d];
    }
    {                      // m_i columns 128..143
        int r = tid >> 4, m = tid & 15;
        sX[r][DIM + m] = mI[(row0 + r) * MD + m];
    }
    if (tid < 64) sX[tid >> 2][144 + (tid & 3)] = 0.f;   // K-pad to 148
    __syncthreads();

    // GEMM1: [16x144] @ [144x256] -> sY (SiLU)
    for (int ct = wave; ct < 16; ct += 8) {
        v8f acc = {0.f,0.f,0.f,0.f,0.f,0.f,0.f,0.f};
        int col = ct * 16 + rc;
        for (int k = 0; k < 144; k += 4) {
            int d0 = k + 2 * half;
            v2f a; a.x = sX[rc][d0]; a.y = sX[rc][d0 + 1];
            v2f b; b.x = nw1g[d0 * 256 + col]; b.y = nw1g[(d0 + 1) * 256 + col];
            acc = wmma4(a, b, acc);
        }
        float bias = nb1g[col];
        for (int v = 0; v < 8; ++v)
            sY[8 * half + v][col] = silu_f(acc[v] + bias);
    }
    __syncthreads();

    // GEMM2: [16x256] @ [256x128] + nb2 + residual
    {
        int ct = wave;
        int col = ct * 16 + rc;
        v8f acc = {0.f,0.f,0.f,0.f,0.f,0.f,0.f,0.f};
        for (int k = 0; k < 256; k += 4) {
            int d0 = k + 2 * half;
            v2f a; a.x = sY[rc][d0]; a.y = sY[rc][d0 + 1];
            v2f b; b.x = nw2g[d0 * DIM + col]; b.y = nw2g[(d0 + 1) * DIM + col];
            acc = wmma4(a, b, acc);
        }
        float bias = nb2g[col];
        for (int v = 0; v < 8; ++v) {
            int row = 8 * half + v;
            fout[(row0 + row) * DIM + col] = acc[v] + bias + sF[row][col];
        }
    }
}

// ---------------- kernel 4: pooled head ----------------
__global__ void k_head(const float* __restrict__ feats,
                       const float* __restrict__ fc1w, const float* __restrict__ fc1b,
                       const float* __restrict__ fc2w, const float* __restrict__ fc2b,
                       float* __restrict__ out) {
    __shared__ float sG[BB * DIM];
    __shared__ float sS[BB * DIM];
    int t = threadIdx.x;   // 0..127
    for (int b = 0; b < BB; ++b) {
        float s = 0.f;
        for (int n = 0; n < NN; ++n) s += feats[(b * NN + n) * DIM + t];
        sG[b * DIM + t] = s * (1.f / NN);
    }
    __syncthreads();
    for (int b = 0; b < BB; ++b) {
        float s = fc1b[t];
        for (int d = 0; d < DIM; ++d) s += sG[b * DIM + d] * fc1w[d * DIM + t];
        sS[b * DIM + t] = silu_f(s);
    }
    __syncthreads();
    if (t < BB * 9) {
        int b = t / 9, k = t % 9;
        float s = fc2b[k];
        for (int d = 0; d < DIM; ++d) s += sS[b * DIM + d] * fc2w[d * 9 + k];
        out[t] = s;
    }
}

// ---------------- host ----------------
extern "C" void kernel_launch(void* const* d_in, const int* in_sizes, int n_in,
                              void* d_out, int out_size, void* d_ws, size_t ws_size,
                              hipStream_t stream) {
    (void)in_sizes; (void)n_in; (void)out_size; (void)ws_size;
    const int*   ids    = (const int*)d_in[0];
    const float* coors  = (const float*)d_in[1];
    const float* tok    = (const float*)d_in[3];
    const float* ew1    = (const float*)d_in[4];
    const float* eb1    = (const float*)d_in[5];
    const float* ew2    = (const float*)d_in[6];
    const float* eb2    = (const float*)d_in[7];
    const float* cw1    = (const float*)d_in[8];
    const float* cb1    = (const float*)d_in[9];
    const float* cw2    = (const float*)d_in[10];
    const float* cb2    = (const float*)d_in[11];
    const float* lng    = (const float*)d_in[12];
    const float* lnb    = (const float*)d_in[13];
    const float* nw1    = (const float*)d_in[14];
    const float* nb1    = (const float*)d_in[15];
    const float* nw2    = (const float*)d_in[16];
    const float* nb2    = (const float*)d_in[17];
    const float* fc1w   = (const float*)d_in[18];
    const float* fc1b   = (const float*)d_in[19];
    const float* fc2w   = (const float*)d_in[20];
    const float* fc2b   = (const float*)d_in[21];
    float* outp = (float*)d_out;

    float* ws = (float*)d_ws;
    float* fA  = ws;  ws += NROW * DIM;
    float* fB  = ws;  ws += NROW * DIM;
    float* cA  = ws;  ws += NROW * 3;
    float* cB  = ws;  ws += NROW * 3;
    float* Fi  = ws;  ws += NROW * FSTR;
    float* Fj  = ws;  ws += NROW * FSTR;
    float* mI  = ws;  ws += NROW * MD;
    float* mIp = ws;  ws += (size_t)JSPL * NROW * MD;
    float* cPp = ws;  ws += (size_t)JSPL * NROW * 3;

    k_embed<<<dim3(NROW), 128, 0, stream>>>(ids, coors, tok, fA, cA);

    for (int l = 0; l < DEPTH; ++l) {
        const float* fin  = (l & 1) ? fB : fA;
        float*       fout = (l & 1) ? fA : fB;
        const float* cin  = (l & 1) ? cB : cA;
        float*       cout = (l & 1) ? cA : cB;
        const float* ew1l = ew1 + (size_t)l * 257 * E2;

        k_node_pre<<<dim3(NROW / 16, 33, 2), 32, 0, stream>>>(
            fin, ew1l, eb1 + l * E2, Fi, Fj);

        k_edge<<<dim3(NROW / 16, JSPL), 256, 0, stream>>>(
            Fi, Fj, cin,
            ew1l + 256 * E2,            // dist weight row
            ew2 + (size_t)l * E2 * MD, eb2 + l * MD,
            cw1 + l * MD * 64, cb1 + l * 64, cw2 + l * 64, cb2 + l,
            mIp, cPp);

        k_combine<<<dim3(64), 256, 0, stream>>>(mIp, cPp, cin, mI, cout);

        k_node_upd<<<dim3(NROW / 16), 256, 0, stream>>>(
            fin, mI, lng + l * DIM, lnb + l * DIM,
            nw1 + (size_t)l * 144 * 256, nb1 + l * 256,
            nw2 + (size_t)l * 256 * DIM, nb2 + l * DIM, fout);
    }

    // DEPTH=3 -> final feats live in fB
    k_head<<<dim3(1), 128, 0, stream>>>(fB, fc1w, fc1b, fc2w, fc2b, outp);
}